// ACmix_5995774345636
// MI455X (gfx1250) — compile-verified
//
#include <hip/hip_runtime.h>
#include <cstdint>

// ---------------------------------------------------------------------------
// ACmix on MI455X (gfx1250, wave32).
//  Kernel 1: qkv = w_qkv[384x128] @ x[b][128x4096]      (WMMA f32, async->LDS)
//  Kernel 2: fused 3x3 window attention + dynamic conv  (VALU, coalesced)
//  Kernel 3: out = w_proj[128x128] @ mix[b][128x4096]   (WMMA f32, async->LDS)
// Workspace: qkv (48MiB) then mix (16MiB).
// ---------------------------------------------------------------------------

typedef __attribute__((ext_vector_type(2))) float v2f;
typedef __attribute__((ext_vector_type(8))) float v8f;

// LDS row pitches chosen for 64-bank conflict freedom (see analysis):
//  A: ds_load_b64 fragments, banks 4r+e..+3 over 32 lanes -> 64 distinct banks.
//  B: ds_load_b32 fragments, 2*72 mod 64 = 16 -> lane halves in disjoint windows.
#define LDS_A_STRIDE 132
#define LDS_B_STRIDE 72

__device__ __forceinline__ unsigned lds_off(const void* p) {
    // flat LDS aperture: addr[31:0] is the LDS byte offset
    return (unsigned)(uintptr_t)p;
}

// gfx1250 async copy global -> LDS (ASYNCcnt), saddr form, 16B per lane.
__device__ __forceinline__ void async_ld_b128(unsigned ldsoff, int voff, const float* base) {
    asm volatile("global_load_async_to_lds_b128 %0, %1, %2"
                 :: "v"(ldsoff), "v"(voff), "s"(base) : "memory");
}

__device__ __forceinline__ void wait_async0() {
#if __has_builtin(__builtin_amdgcn_s_wait_asynccnt)
    __builtin_amdgcn_s_wait_asynccnt(0);
#else
    asm volatile("s_wait_asynccnt 0x0" ::: "memory");
#endif
}

// Out[M,4096] = A[M,128] * B[128,4096] for batch blockIdx.z (A shared).
// Block = 128 threads = 4 waves; block tile 64M x 64N, full K=128 staged once.
__global__ void __launch_bounds__(128)
wmma_gemm_f32(const float* __restrict__ A, const float* __restrict__ Bm,
              float* __restrict__ Out, int N, long strideB, long strideOut) {
    constexpr int KT = 128;
    __shared__ float sA[64 * LDS_A_STRIDE];   // 33 KB
    __shared__ float sB[KT * LDS_B_STRIDE];   // 36 KB

    Bm  += (size_t)blockIdx.z * strideB;
    Out += (size_t)blockIdx.z * strideOut;
    const int m0 = blockIdx.y * 64;
    const int n0 = blockIdx.x * 64;
    const int tid = threadIdx.x;

    // ---- async-stage A(64x128, contiguous) and B(128 rows x 64 of N) ----
    const float* Ablk = A + (size_t)m0 * KT;
    const float* Bblk = Bm + n0;
    #pragma unroll
    for (int i = 0; i < 4; i++) {
        const int c = tid + i * 128;          // chunk id 0..511 (16B chunks)
        const int ma = c >> 3, qa = c & 7;    // A: 8 chunks per 128-float row
        async_ld_b128(lds_off(&sA[ma * LDS_A_STRIDE + qa * 4]),
                      (ma * KT + qa * 4) * 4, Ablk);
        const int kb = c >> 2, qb = c & 3;    // B: 4 chunks per 64-float row
        async_ld_b128(lds_off(&sB[kb * LDS_B_STRIDE + qb * 4]),
                      (kb * N + qb * 4) * 4, Bblk);
    }
    wait_async0();
    __syncthreads();

    // ---- WMMA from LDS: wave = 16M x 64N tile ----
    const int wave = tid >> 5;
    const int lane = tid & 31;
    const int lh   = lane & 15;
    const int koff = (lane >> 4) << 1;        // 0 or 2

    v8f acc0 = {}, acc1 = {}, acc2 = {}, acc3 = {};
    const float* aP = &sA[(wave * 16 + lh) * LDS_A_STRIDE + koff];
    const float* bP = &sB[koff * LDS_B_STRIDE + lh];

    #pragma unroll 4
    for (int k = 0; k < KT; k += 4) {
        v2f a = *(const v2f*)(aP + k);                   // ds_load_b64
        const float* b0 = bP + k * LDS_B_STRIDE;
        const float* b1 = b0 + LDS_B_STRIDE;
        v2f f0 = {b0[0],  b1[0]};                        // all frags loaded
        v2f f1 = {b0[16], b1[16]};                       // before the WMMAs
        v2f f2 = {b0[32], b1[32]};
        v2f f3 = {b0[48], b1[48]};
        acc0 = __builtin_amdgcn_wmma_f32_16x16x4_f32(false, a, false, f0, (short)0, acc0, false, false);
        acc1 = __builtin_amdgcn_wmma_f32_16x16x4_f32(false, a, false, f1, (short)0, acc1, false, false);
        acc2 = __builtin_amdgcn_wmma_f32_16x16x4_f32(false, a, false, f2, (short)0, acc2, false, false);
        acc3 = __builtin_amdgcn_wmma_f32_16x16x4_f32(false, a, false, f3, (short)0, acc3, false, false);
    }

    // C/D layout: VGPR i = row m0w + (lane>>4)*8 + i, col = n0 + j*16 + (lane&15)
    const int rowBase = m0 + wave * 16 + ((lane >> 4) << 3);
    float* obase = Out + (size_t)rowBase * N + n0 + lh;
    #pragma unroll
    for (int i = 0; i < 8; i++) {
        float* o = obase + (size_t)i * N;
        o[0]  = acc0[i];
        o[16] = acc1[i];
        o[32] = acc2[i];
        o[48] = acc3[i];
    }
}

// Fused: window attention (softmax over 3x3) + dynamic grouped conv + alpha/beta mix.
// One thread per (batch,pixel). All k/v gathers are "l + di*W + dj" -> coalesced.
__global__ void __launch_bounds__(256)
acmix_middle(const float* __restrict__ qkv, const float* __restrict__ w_kernel,
             const float* __restrict__ b_kernel, const float* __restrict__ alpha_p,
             const float* __restrict__ beta_p, float* __restrict__ mix) {
    constexpr int C = 128, H = 64, W = 64, L = H * W, HD = 32, K2 = 9, CG = 96;
    const int idx = blockIdx.x * blockDim.x + threadIdx.x;   // b*L + l (grid exact)
    const int b = idx / L;
    const int l = idx - b * L;
    const int y = l / W;
    const int x = l - y * W;

    const float alpha = alpha_p[0];
    const float beta  = beta_p[0];

    const float* qb = qkv + (size_t)b * 3 * C * L;
    const float* kb = qb + (size_t)C * L;
    const float* vb = qb + (size_t)2 * C * L;

    int   noff[K2];
    float nmask[K2];
    #pragma unroll
    for (int t = 0; t < K2; t++) {
        const int di = t / 3 - 1, dj = t % 3 - 1;
        const int yy = y + di, xx = x + dj;
        const bool ok = (yy >= 0) & (yy < H) & (xx >= 0) & (xx < W);
        noff[t]  = ok ? (yy * W + xx) : 0;
        nmask[t] = ok ? 1.0f : 0.0f;
    }

    float wmix[4][K2];

    for (int h = 0; h < 4; h++) {
        float logits[K2];
        #pragma unroll
        for (int t = 0; t < K2; t++) logits[t] = 0.0f;
        const float* qrow = qb + (size_t)(h * HD) * L;
        const float* krow = kb + (size_t)(h * HD) * L;
        for (int d = 0; d < HD; d++) {
            const float qv = qrow[(size_t)d * L + l];
            const float* kr = krow + (size_t)d * L;
            #pragma unroll
            for (int t = 0; t < K2; t++)
                logits[t] += qv * (kr[noff[t]] * nmask[t]);
        }
        float m = logits[0];
        #pragma unroll
        for (int t = 1; t < K2; t++) m = fmaxf(m, logits[t]);
        float s = 0.0f;
        float e[K2];
        #pragma unroll
        for (int t = 0; t < K2; t++) { e[t] = expf(logits[t] - m); s += e[t]; }
        const float inv = alpha / s;
        #pragma unroll
        for (int t = 0; t < K2; t++) wmix[h][t] = e[t] * inv;
    }

    for (int g = 0; g < 4; g++) {
        float dynk[K2];
        #pragma unroll
        for (int t = 0; t < K2; t++) dynk[t] = b_kernel[g * K2 + t];
        const float* qg  = qb + (size_t)(g * CG) * L + l;
        const float* wkg = w_kernel + (size_t)g * K2 * CG;
        for (int c = 0; c < CG; c++) {
            const float v = qg[(size_t)c * L];
            #pragma unroll
            for (int t = 0; t < K2; t++) dynk[t] += v * wkg[(size_t)t * CG + c];
        }
        #pragma unroll
        for (int t = 0; t < K2; t++) wmix[g][t] += beta * dynk[t];
    }

    float* mout = mix + (size_t)b * C * L + l;
    for (int h = 0; h < 4; h++) {
        const float* vrow = vb + (size_t)(h * HD) * L;
        for (int d = 0; d < HD; d++) {
            const float* vr = vrow + (size_t)d * L;
            float acc = 0.0f;
            #pragma unroll
            for (int t = 0; t < K2; t++)
                acc += wmix[h][t] * (vr[noff[t]] * nmask[t]);
            mout[(size_t)(h * HD + d) * L] = acc;
        }
    }
}

extern "C" void kernel_launch(void* const* d_in, const int* in_sizes, int n_in,
                              void* d_out, int out_size, void* d_ws, size_t ws_size,
                              hipStream_t stream) {
    const float* x        = (const float*)d_in[0];  // [8,128,64,64]
    const float* w_qkv    = (const float*)d_in[1];  // [384,128]
    const float* w_kernel = (const float*)d_in[2];  // [36,96]
    const float* b_kernel = (const float*)d_in[3];  // [36]
    const float* w_proj   = (const float*)d_in[4];  // [128,128]
    const float* alpha    = (const float*)d_in[5];  // [1]
    const float* beta     = (const float*)d_in[6];  // [1]
    float* out = (float*)d_out;                     // [8,128,64,64]

    const int B = 8, C = 128, L = 64 * 64;
    float* qkv_ws = (float*)d_ws;                           // 8*384*4096 floats
    float* mix_ws = qkv_ws + (size_t)B * 3 * C * L;         // 8*128*4096 floats

    dim3 blk(128);
    // qkv GEMM: [384x128] @ x[b][128x4096]
    wmma_gemm_f32<<<dim3(L / 64, (3 * C) / 64, B), blk, 0, stream>>>(
        w_qkv, x, qkv_ws, L, (long)C * L, (long)3 * C * L);

    // fused window attention + dynamic conv
    acmix_middle<<<(B * L) / 256, 256, 0, stream>>>(
        qkv_ws, w_kernel, b_kernel, alpha, beta, mix_ws);

    // projection GEMM: [128x128] @ mix[b][128x4096]
    wmma_gemm_f32<<<dim3(L / 64, C / 64, B), blk, 0, stream>>>(
        w_proj, mix_ws, out, L, (long)C * L, (long)C * L);
}